// ConceptEmbedModel_65695819759691
// MI455X (gfx1250) — compile-verified
//
#include <hip/hip_runtime.h>

#define DDIM 200            // embedding dim (fixed in reference)
#define TILE 16

typedef float v2f __attribute__((ext_vector_type(2)));
typedef float v8f __attribute__((ext_vector_type(8)));

// ---------------------------------------------------------------------------
// Kernel 1: deterministic ragged segment-sum.
// segment_ids is sorted, so block c binary-searches its [start,end) range and
// accumulates embed rows in registers. No atomics -> bitwise deterministic.
// ---------------------------------------------------------------------------
__global__ void __launch_bounds__(256)
segsum_kernel(const float* __restrict__ embed,
              const int*   __restrict__ aids,
              const int*   __restrict__ segs,
              int T,
              float* __restrict__ H)
{
    const int c = blockIdx.x;

    // lower_bound(segs, c)
    int lo = 0, hi = T;
    while (lo < hi) { int mid = (lo + hi) >> 1; if (segs[mid] <  c) lo = mid + 1; else hi = mid; }
    const int start = lo;
    // upper_bound(segs, c)
    hi = T;
    while (lo < hi) { int mid = (lo + hi) >> 1; if (segs[mid] <= c) lo = mid + 1; else hi = mid; }
    const int end = lo;

    const int d = threadIdx.x;
    if (d < DDIM) {
        float acc = 0.0f;
        for (int t = start; t < end; ++t) {
            const int a = aids[t];                 // uniform across block -> scalarizable
            acc += embed[(long long)a * DDIM + d];
        }
        H[(long long)c * DDIM + d] = acc;
    }
}

// ---------------------------------------------------------------------------
// Kernel 2: logits = x [B,200] @ H^T [C,200] using V_WMMA_F32_16X16X4_F32.
// One wave (32 lanes) owns one 16x16 output tile; K loop of 50 x (K=4) WMMAs.
// A  (16x4 f32): lane = m + 16*(k>=2), vgpr = k&1      -> b64 load from x
// B  (4x16 f32): lane = n + 16*(k>=2), vgpr = k&1      -> b64 load from H
// C/D (16x16 f32, 8 VGPRs): row M = v + 8*(lane>=16), col N = lane&15
// ---------------------------------------------------------------------------
__global__ void __launch_bounds__(256)
gemm_wmma_kernel(const float* __restrict__ x,
                 const float* __restrict__ H,
                 float* __restrict__ out,
                 int rowTiles, int colTiles, int C)
{
    const int wave = blockIdx.x * (blockDim.x >> 5) + (threadIdx.x >> 5);
    if (wave >= rowTiles * colTiles) return;       // wave-uniform: EXEC stays all-1s

    const int lane = threadIdx.x & 31;
    const int rowTile = wave / colTiles;
    const int colTile = wave % colTiles;

    const int m0   = rowTile * TILE;
    const int n0   = colTile * TILE;
    const int half = lane >> 4;                    // 0: K=0..1, 1: K=2..3
    const int l15  = lane & 15;

    const float* __restrict__ xrow = x + (long long)(m0 + l15) * DDIM + half * 2;
    const float* __restrict__ hrow = H + (long long)(n0 + l15) * DDIM + half * 2;

    v8f acc = {};
    #pragma unroll 5
    for (int k0 = 0; k0 < DDIM; k0 += 4) {
        v2f a = *(const v2f*)(xrow + k0);
        v2f b = *(const v2f*)(hrow + k0);
        // (neg_a, A, neg_b, B, c_mod, C, reuse_a, reuse_b)
        acc = __builtin_amdgcn_wmma_f32_16x16x4_f32(
                  false, a, false, b, (short)0, acc, false, false);
    }

    float* __restrict__ o = out + (long long)(m0 + half * 8) * C + n0 + l15;
    #pragma unroll
    for (int v = 0; v < 8; ++v)
        o[(long long)v * C] = acc[v];
}

// ---------------------------------------------------------------------------
// Kernel 3: in-place row softmax over C elements. One block per row.
// Online (max,sum) per thread, LDS tree combine, then normalize pass.
// ---------------------------------------------------------------------------
__global__ void __launch_bounds__(256)
softmax_kernel(float* __restrict__ out, int C)
{
    __shared__ float sm[256];
    __shared__ float ss[256];

    float* __restrict__ row = out + (long long)blockIdx.x * C;
    const int tid = threadIdx.x;

    float m = -__builtin_inff();
    float s = 0.0f;
    for (int i = tid; i < C; i += 256) {
        const float v = row[i];
        if (v > m) { s = s * __expf(m - v) + 1.0f; m = v; }
        else       { s += __expf(v - m); }
    }
    sm[tid] = m; ss[tid] = s;
    __syncthreads();

    for (int off = 128; off > 0; off >>= 1) {
        if (tid < off) {
            const float m1 = sm[tid],       s1 = ss[tid];
            const float m2 = sm[tid + off], s2 = ss[tid + off];
            const float M  = fmaxf(m1, m2);
            ss[tid] = s1 * __expf(m1 - M) + s2 * __expf(m2 - M);
            sm[tid] = M;
        }
        __syncthreads();
    }

    const float M   = sm[0];
    const float inv = 1.0f / ss[0];
    for (int i = tid; i < C; i += 256)
        row[i] = __expf(row[i] - M) * inv;
}

// ---------------------------------------------------------------------------
// Inputs (setup_inputs order): x[B*200] f32, embed_weight[400000*200] f32,
// ancestor_ids[T] i32, segment_ids[T] i32, n_concepts (scalar, unused on host
// side — C recovered from out_size). Output: softmax [B, C] f32 in d_out.
// Workspace: local_H [C, 200] f32 at d_ws offset 0 (40 MB).
// ---------------------------------------------------------------------------
extern "C" void kernel_launch(void* const* d_in, const int* in_sizes, int n_in,
                              void* d_out, int out_size, void* d_ws, size_t ws_size,
                              hipStream_t stream)
{
    const float* x     = (const float*)d_in[0];
    const float* embed = (const float*)d_in[1];
    const int*   aids  = (const int*)d_in[2];
    const int*   segs  = (const int*)d_in[3];

    const int T = in_sizes[2];
    const int B = in_sizes[0] / DDIM;     // 1024
    const int C = out_size / B;           // 50000

    float* H   = (float*)d_ws;            // [C, 200]
    float* out = (float*)d_out;           // [B, C]

    // Stage 1: local_H = segment_sum(embed[aids], segs)
    segsum_kernel<<<C, 256, 0, stream>>>(embed, aids, segs, T, H);

    // Stage 2: logits = x @ H^T via fp32 WMMA
    const int rowTiles = B / TILE;        // 64
    const int colTiles = C / TILE;        // 3125
    const long long waves = (long long)rowTiles * colTiles;
    const int wavesPerBlock = 256 / 32;   // 8 waves / block
    const int blocks = (int)((waves + wavesPerBlock - 1) / wavesPerBlock);
    gemm_wmma_kernel<<<blocks, 256, 0, stream>>>(x, H, out, rowTiles, colTiles, C);

    // Stage 3: in-place row softmax
    softmax_kernel<<<B, 256, 0, stream>>>(out, C);
}